// ConvNet_26792005992530
// MI455X (gfx1250) — compile-verified
//
#include <hip/hip_runtime.h>
#include <stdint.h>

// ---------------------------------------------------------------------------
// CDNA5 (gfx1250) fp32 WMMA implementation of:
//   conv7x7/s3 (3->4) -> square -> FC 324->64 (+b) -> square -> FC 64->10 (+b)
// All GEMMs use V_WMMA_F32_16X16X4_F32 (full fp32 path, matches fp32 ref).
// One wave owns one 16x16 C tile; wave32, EXEC uniform through all WMMA loops.
// Conv A-operand addressing: uniform SGPR base + 32-bit per-lane byte offset
// (x is 201 MB < 2^31) so each gather is one v_add_nc_u32 + saddr global load.
// ---------------------------------------------------------------------------

typedef __attribute__((ext_vector_type(2))) float v2f;
typedef __attribute__((ext_vector_type(8))) float v8f;

#define C_IN    3
#define IMG     32
#define C_OUT   4
#define KS      7
#define STR     3
#define HO      9
#define NPIX    81           // 9*9 output pixels
#define KCONV   147          // 3*7*7
#define KCONV_P 148          // padded to multiple of 4 (37 wmma steps)
#define FC1_IN  324          // 4*81, exactly 81 wmma steps
#define HIDDEN  64
#define OUTPUT  10

__device__ __forceinline__ v8f wmma4(v2f a, v2f b, v8f c) {
  // 8 args: (neg_a, A, neg_b, B, c_mod, C, reuse_a, reuse_b)
  return __builtin_amdgcn_wmma_f32_16x16x4_f32(false, a, false, b, (short)0, c,
                                               false, false);
}

__device__ __forceinline__ float ldg_off(const float* __restrict__ base,
                                         uint32_t byte_off) {
  // uniform base (SGPR pair) + zext(32-bit byte offset) -> saddr global load
  return *(const float*)((const char*)base + (size_t)byte_off);
}

// ---------------------------------------------------------------------------
// Kernel 1: conv (im2col GEMM, N padded 4->16) + square.
//   y[b*324 + co*81 + p] = (sum_k x_im2col[(b,p), k] * w[co, k])^2
// Block = 256 threads = 8 waves; each wave: one 16-row M tile, full K loop.
// ---------------------------------------------------------------------------
__global__ __launch_bounds__(256)
void conv_sq_kernel(const float* __restrict__ x, const float* __restrict__ w,
                    float* __restrict__ y, int nrows) {
  __shared__ float wT[16 * KCONV_P];   // wT[n][k], zero-padded (n>=4 or k>=147)
  __shared__ int   offs[KCONV_P];      // im2col k -> BYTE offset within window

  const int tid = threadIdx.x;
  for (int idx = tid; idx < 16 * KCONV_P; idx += 256) {
    int n = idx / KCONV_P, k = idx - n * KCONV_P;
    wT[idx] = (n < C_OUT && k < KCONV) ? w[n * KCONV + k] : 0.0f;
  }
  for (int idx = tid; idx < KCONV_P; idx += 256) {
    int k = (idx < KCONV) ? idx : (KCONV - 1);   // pad col: weight is 0 anyway
    int ci = k / 49, r = k - ci * 49;
    int ky = r / 7,  kx = r - ky * 7;
    offs[idx] = (ci * (IMG * IMG) + ky * IMG + kx) * 4;   // bytes
  }
  __syncthreads();

  const int lane = tid & 31;
  const int wave = tid >> 5;
  const int ln   = lane & 15;
  const int half = lane >> 4;
  const int koff = half * 2;

  const int row_base = (blockIdx.x * 8 + wave) * 16;
  if (row_base >= nrows) return;

  // A-fragment source: this lane feeds im2col row (row_base + ln).
  // Per-lane 32-bit byte offset of the conv window origin inside x.
  const int arow = row_base + ln;
  const int b    = arow / NPIX;
  const int p    = arow - b * NPIX;
  const int oy   = p / HO, ox = p - oy * HO;
  const uint32_t base =
      ((uint32_t)b * (C_IN * IMG * IMG) +
       (uint32_t)(oy * STR * IMG + ox * STR)) * 4u;

  v8f acc = {};
  for (int kt = 0; kt < KCONV_P / 4; ++kt) {
    const int k0 = kt * 4 + koff;
    v2f a, bw;
    a.x = ldg_off(x, base + (uint32_t)offs[k0]);
    a.y = ldg_off(x, base + (uint32_t)offs[k0 + 1]);
    bw  = *(const v2f*)&wT[ln * KCONV_P + k0];   // 8B-aligned: k0 even
    acc = wmma4(a, bw, acc);
  }

  // Epilogue: C layout -> lanes 0-15 hold M=r, lanes 16-31 hold M=r+8; N=ln.
  if (ln < C_OUT) {
#pragma unroll
    for (int r = 0; r < 8; ++r) {
      const int row = row_base + half * 8 + r;
      const int bb  = row / NPIX;
      const int pp  = row - bb * NPIX;
      const float c = acc[r];
      y[(size_t)bb * FC1_IN + ln * NPIX + pp] = c * c;
    }
  }
}

// ---------------------------------------------------------------------------
// Kernel 2: h = (y @ fc1_w^T + b)^2.  M=B, K=324 (81 steps), N=64.
// 8 waves/block: 2 M-tiles x 4 N-tiles (32 rows x 64 cols per block).
// ---------------------------------------------------------------------------
__global__ __launch_bounds__(256)
void fc1_sq_kernel(const float* __restrict__ y, const float* __restrict__ w,
                   const float* __restrict__ bias, float* __restrict__ h) {
  const int tid  = threadIdx.x;
  const int lane = tid & 31;
  const int wave = tid >> 5;
  const int ln   = lane & 15;
  const int half = lane >> 4;
  const int koff = half * 2;

  const int m_base = blockIdx.x * 32 + (wave >> 2) * 16;
  const int n      = (wave & 3) * 16 + ln;

  const float* ap = y + (size_t)(m_base + ln) * FC1_IN;  // row stride even
  const float* bp = w + (size_t)n * FC1_IN;

  v8f acc = {};
  for (int kt = 0; kt < FC1_IN / 4; ++kt) {
    const int k0 = kt * 4 + koff;                 // even -> 8B aligned
    const v2f a = *(const v2f*)(ap + k0);
    const v2f b = *(const v2f*)(bp + k0);
    acc = wmma4(a, b, acc);
  }

  const float bn = bias[n];
#pragma unroll
  for (int r = 0; r < 8; ++r) {
    const int row = m_base + half * 8 + r;
    const float t = acc[r] + bn;
    h[(size_t)row * HIDDEN + n] = t * t;
  }
}

// ---------------------------------------------------------------------------
// Kernel 3: out = h @ fc2_w^T + b.  M=B, K=64 (16 steps), N=10 padded to 16.
// Weights zero-padded in LDS so the WMMA loop stays EXEC-uniform.
// ---------------------------------------------------------------------------
__global__ __launch_bounds__(256)
void fc2_kernel(const float* __restrict__ h, const float* __restrict__ w,
                const float* __restrict__ bias, float* __restrict__ out) {
  __shared__ float wpad[16 * HIDDEN];  // wpad[n][k], zero for n >= 10
  const int tid = threadIdx.x;
  for (int idx = tid; idx < 16 * HIDDEN; idx += 256) {
    int n = idx / HIDDEN, k = idx - n * HIDDEN;
    wpad[idx] = (n < OUTPUT) ? w[n * HIDDEN + k] : 0.0f;
  }
  __syncthreads();

  const int lane = tid & 31;
  const int wave = tid >> 5;
  const int ln   = lane & 15;
  const int half = lane >> 4;
  const int koff = half * 2;

  const int m_base = (blockIdx.x * 8 + wave) * 16;
  const float* ap  = h + (size_t)(m_base + ln) * HIDDEN;

  v8f acc = {};
  for (int kt = 0; kt < HIDDEN / 4; ++kt) {
    const int k0 = kt * 4 + koff;                 // even -> 8B aligned
    const v2f a = *(const v2f*)(ap + k0);
    const v2f b = *(const v2f*)&wpad[ln * HIDDEN + k0];
    acc = wmma4(a, b, acc);
  }

  if (ln < OUTPUT) {
    const float bn = bias[ln];
#pragma unroll
    for (int r = 0; r < 8; ++r) {
      const int row = m_base + half * 8 + r;
      out[(size_t)row * OUTPUT + ln] = acc[r] + bn;
    }
  }
}

// ---------------------------------------------------------------------------
extern "C" void kernel_launch(void* const* d_in, const int* in_sizes, int n_in,
                              void* d_out, int out_size, void* d_ws,
                              size_t ws_size, hipStream_t stream) {
  const float* x      = (const float*)d_in[0];
  const float* conv_w = (const float*)d_in[1];
  const float* fc1_w  = (const float*)d_in[2];
  const float* fc1_b  = (const float*)d_in[3];
  const float* fc2_w  = (const float*)d_in[4];
  const float* fc2_b  = (const float*)d_in[5];
  float* out = (float*)d_out;

  const int B = in_sizes[0] / (C_IN * IMG * IMG);  // 16384

  float* y = (float*)d_ws;                  // [B, 324] squared conv output
  float* h = y + (size_t)B * FC1_IN;        // [B, 64]  squared FC1 output
  // workspace used: B*(324+64)*4 bytes ~= 25.4 MB

  const int nrows = B * NPIX;               // 1,327,104 im2col rows
  conv_sq_kernel<<<nrows / (16 * 8), 256, 0, stream>>>(x, conv_w, y, nrows);
  fc1_sq_kernel<<<B / 32, 256, 0, stream>>>(y, fc1_w, fc1_b, h);
  fc2_kernel<<<B / (16 * 8), 256, 0, stream>>>(h, fc2_w, fc2_b, out);
}